// SGConv_52613349376206
// MI455X (gfx1250) — compile-verified
//
#include <hip/hip_runtime.h>
#include <hip/hip_bf16.h>
#include <math.h>

// Problem constants (match the reference harness).
#define NNODES 16384   // N
#define DFEAT  128     // D

typedef __attribute__((ext_vector_type(2))) float v2f;
typedef __attribute__((ext_vector_type(8))) float v8f;

// ---------------------------------------------------------------------------
// 1) deg[b*N + row] += |edge_val|
// ---------------------------------------------------------------------------
__global__ void sg_deg_kernel(const int* __restrict__ eb, const int* __restrict__ er,
                              const float* __restrict__ ev, float* __restrict__ deg, int E) {
  int i = blockIdx.x * 256 + threadIdx.x;
  if (i < E) {
    atomicAdd(&deg[(size_t)eb[i] * NNODES + er[i]], fabsf(ev[i]));
  }
}

// ---------------------------------------------------------------------------
// 2) norm = (deg + 1e-6)^-0.5
// ---------------------------------------------------------------------------
__global__ void sg_norm_kernel(const float* __restrict__ deg, float* __restrict__ nrm, int BN) {
  int i = blockIdx.x * 256 + threadIdx.x;
  if (i < BN) nrm[i] = rsqrtf(deg[i] + 1e-6f);
}

// ---------------------------------------------------------------------------
// 3) y = (x @ W) * norm[row]   via V_WMMA_F32_16X16X4_F32
//    grid.x = (B*N)/16 row-blocks; 256 threads = 8 waves; wave w -> col tile w.
//    Fragment layouts (ISA 7.12.2, wave32):
//      A 16x4 : lanes 0-15 row M=lane,   VGPR0/1 = K=0/1 ; lanes 16-31 K=2/3
//      B 4x16 : lanes 0-15 col N=lane,   VGPR0/1 = K=0/1 ; lanes 16-31 K=2/3
//      C/D    : VGPR i, lanes 0-15 -> (M=i,   N=lane)
//                         lanes 16-31 -> (M=i+8, N=lane-16)
// ---------------------------------------------------------------------------
__global__ void sg_gemm_norm_kernel(const float* __restrict__ x, const float* __restrict__ W,
                                    const float* __restrict__ nrm, float* __restrict__ y) {
  const int wave = threadIdx.x >> 5;   // 0..7 -> column tile
  const int lane = threadIdx.x & 31;
  const int lh   = lane >> 4;          // lane-half: 0 => K+{0,1}, 1 => K+{2,3}
  const int ll   = lane & 15;
  const int row0 = blockIdx.x << 4;    // 16 rows of x per block
  const int col0 = wave << 4;          // 16 cols of W per wave

  // A fragment source: x[row0+ll, k + 2*lh + {0,1}]  (two adjacent floats -> b64 load)
  const float* xrow = x + (size_t)(row0 + ll) * DFEAT + 2 * lh;
  // B fragment source: W[k + 2*lh + {0,1}, col0+ll]  (stride-D apart)
  const float* wcol = W + (size_t)(2 * lh) * DFEAT + (col0 + ll);

  v8f c = {};
#pragma unroll
  for (int k = 0; k < DFEAT; k += 4) {
    v2f a = *(const v2f*)(xrow + k);           // K=k+2lh, k+2lh+1
    v2f bm;
    bm.x = wcol[(size_t)k * DFEAT];            // K=k+2lh   row of W
    bm.y = wcol[(size_t)k * DFEAT + DFEAT];    // K=k+2lh+1 row of W
    c = __builtin_amdgcn_wmma_f32_16x16x4_f32(
        /*neg_a=*/false, a, /*neg_b=*/false, bm,
        /*c_mod=*/(short)0, c, /*reuse_a=*/false, /*reuse_b=*/false);
  }

  // Epilogue: scale by norm of the element's own row, store y.
#pragma unroll
  for (int i = 0; i < 8; ++i) {
    int r = row0 + i + 8 * lh;
    y[(size_t)r * DFEAT + col0 + ll] = c[i] * nrm[r];
  }
}

// ---------------------------------------------------------------------------
// 4) SpMM: agg[dst] += edge_val * y[src]; one wave per edge, float4 per lane.
// ---------------------------------------------------------------------------
__global__ void sg_spmm_kernel(const float* __restrict__ y,
                               const int* __restrict__ eb, const int* __restrict__ er,
                               const int* __restrict__ ec, const float* __restrict__ ev,
                               float* __restrict__ agg, int E) {
  const int grp  = threadIdx.x >> 5;   // 8 edges per 256-thread block
  const int lane = threadIdx.x & 31;   // 32 lanes * float4 = 128 feats
  const int e = blockIdx.x * 8 + grp;
  if (e >= E) return;
  const int    b = eb[e];
  const int    r = er[e];
  const int    s = ec[e];
  const float  v = ev[e];
  const float4 m = ((const float4*)(y + ((size_t)b * NNODES + s) * DFEAT))[lane];
  float* dst = agg + ((size_t)b * NNODES + r) * DFEAT + lane * 4;
  atomicAdd(dst + 0, m.x * v);
  atomicAdd(dst + 1, m.y * v);
  atomicAdd(dst + 2, m.z * v);
  atomicAdd(dst + 3, m.w * v);
}

// ---------------------------------------------------------------------------
// 5) out = relu(agg * norm[row] + bias[col])
// ---------------------------------------------------------------------------
__global__ void sg_finish_kernel(const float* __restrict__ agg, const float* __restrict__ nrm,
                                 const float* __restrict__ bias, float* __restrict__ out, int BN) {
  const int idx = blockIdx.x * 256 + threadIdx.x;  // one float4 per thread
  const int row = idx >> 5;                        // 32 float4 per row
  const int q   = idx & 31;
  if (row >= BN) return;
  const float  nv = nrm[row];
  const float4 a  = ((const float4*)(agg + (size_t)row * DFEAT))[q];
  const float4 bv = ((const float4*)bias)[q];
  float4 o;
  o.x = fmaxf(fmaf(a.x, nv, bv.x), 0.0f);
  o.y = fmaxf(fmaf(a.y, nv, bv.y), 0.0f);
  o.z = fmaxf(fmaf(a.z, nv, bv.z), 0.0f);
  o.w = fmaxf(fmaf(a.w, nv, bv.w), 0.0f);
  ((float4*)(out + (size_t)row * DFEAT))[q] = o;
}

// ---------------------------------------------------------------------------
extern "C" void kernel_launch(void* const* d_in, const int* in_sizes, int n_in,
                              void* d_out, int out_size, void* d_ws, size_t ws_size,
                              hipStream_t stream) {
  const float* x    = (const float*)d_in[0];   // [B,N,D]
  const float* W    = (const float*)d_in[1];   // [D,D]
  const float* bias = (const float*)d_in[2];   // [D]
  const int*   eb   = (const int*)d_in[3];     // [E]
  const int*   er   = (const int*)d_in[4];     // [E]
  const int*   ec   = (const int*)d_in[5];     // [E]
  const float* ev   = (const float*)d_in[6];   // [E]
  // d_in[7] = k (unused: norm_k is computed but unused in the reference)

  const int E  = in_sizes[3];
  const int BN = in_sizes[0] / DFEAT;          // B*N = 65536

  // Workspace layout (floats): deg[BN] | norm[BN] | y[BN*D] | agg[BN*D]
  float* deg  = (float*)d_ws;
  float* nrm  = deg + BN;
  float* y    = nrm + BN;
  float* agg  = y + (size_t)BN * DFEAT;

  hipMemsetAsync(deg, 0, (size_t)BN * sizeof(float), stream);
  hipMemsetAsync(agg, 0, (size_t)BN * DFEAT * sizeof(float), stream);

  sg_deg_kernel<<<(E + 255) / 256, 256, 0, stream>>>(eb, er, ev, deg, E);
  sg_norm_kernel<<<(BN + 255) / 256, 256, 0, stream>>>(deg, nrm, BN);
  sg_gemm_norm_kernel<<<BN / 16, 256, 0, stream>>>(x, W, nrm, y);
  sg_spmm_kernel<<<(E + 7) / 8, 256, 0, stream>>>(y, eb, er, ec, ev, agg, E);
  sg_finish_kernel<<<(BN * (DFEAT / 4) + 255) / 256, 256, 0, stream>>>(agg, nrm, bias,
                                                                       (float*)d_out, BN);
}